// MPNNModel_A_T_L_68573447848206
// MI455X (gfx1250) — compile-verified
//
#include <hip/hip_runtime.h>
#include <hip/hip_bf16.h>

// ---------------- static config (matches reference) ----------------
#define N0 150000
#define N1 75000
#define NEDGE 200000
#define DIM 128
#define DIMP 132            // padded LDS row stride (floats) -> conflict-free
#define NLAYERS 3
#define NTYPES 4
#define NC 3

typedef __attribute__((ext_vector_type(2))) float v2f;
typedef __attribute__((ext_vector_type(8))) float v8f;

#if defined(__AMDGCN__) && __has_builtin(__builtin_amdgcn_global_load_async_to_lds_b128)
#define HAVE_ASYNC_LDS 1
typedef __attribute__((__vector_size__(16))) int v4i;
typedef __attribute__((address_space(1))) v4i gv4i;   // global AS pointee
typedef __attribute__((address_space(3))) v4i lv4i;   // LDS AS pointee
#else
#define HAVE_ASYNC_LDS 0
#endif

// ---------------- elementwise helpers (float4-vectorized) ----------------
__global__ void fill4_kernel(float4* __restrict__ p, float v, int n4) {
    int i = blockIdx.x * blockDim.x + threadIdx.x;
    if (i < n4) p[i] = make_float4(v, v, v, v);
}

// one wave per row: lane covers 4 contiguous floats
__global__ void gather_kernel(const float* __restrict__ table,
                              const int* __restrict__ ids,
                              float* __restrict__ dst, int nrows) {
    int gid = blockIdx.x * blockDim.x + threadIdx.x;
    int row = gid >> 5, lane = gid & 31;
    if (row >= nrows) return;
    int d = lane * 4;
    *(float4*)(dst + (size_t)row * DIM + d) =
        *(const float4*)(table + (size_t)ids[row] * DIM + d);
}

// out = relu(agg)   (accumulate==0)  or  out += relu(agg)  (accumulate==1)
__global__ void relu_acc_kernel(float4* __restrict__ out,
                                const float4* __restrict__ agg, int n4,
                                int accumulate) {
    int i = blockIdx.x * blockDim.x + threadIdx.x;
    if (i >= n4) return;
    float4 a = agg[i];
    float4 v = make_float4(fmaxf(a.x, 0.f), fmaxf(a.y, 0.f),
                           fmaxf(a.z, 0.f), fmaxf(a.w, 0.f));
    if (accumulate) {
        float4 o = out[i];
        v.x += o.x; v.y += o.y; v.z += o.z; v.w += o.w;
    }
    out[i] = v;
}

// ---------------- fp32 WMMA GEMM: H = X @ W^T + bias ----------------
// Block = 256 threads = 8 waves = 8 column tiles of a 32-row stripe.
// The shared 32x128 A stripe is staged once into LDS via async global->LDS
// (CDNA5 GLOBAL_LOAD_ASYNC_TO_LDS_B128, ASYNCcnt), removing the 8x redundant
// A fetches. Each wave runs two independent 32-deep WMMA chains (32x16 tile)
// sharing each B fragment.
__global__ void __launch_bounds__(256)
gemm_bias_wmma_kernel(const float* __restrict__ X, const float* __restrict__ W,
                      const float* __restrict__ bias, float* __restrict__ H,
                      int N) {
    __shared__ float Atile[32 * DIMP];     // 16.5 KB, padded rows

    const int tid  = threadIdx.x;
    const int wave = tid >> 5;             // 0..7 -> column tile
    const int lane = tid & 31;
    const int lm   = lane & 15;
    const int hi   = lane >> 4;            // 0: K=[k,k+1], 1: K=[k+2,k+3]
    const int row0 = blockIdx.x * 32;
    const int col0 = wave * 16;

    // ---- stage A tile: 32 rows x 128 floats = 1024 x 16B chunks, 4 rounds
#pragma unroll
    for (int r = 0; r < 4; ++r) {
        int c    = r * 256 + tid;
        int trow = c >> 5;                 // 0..31
        int tcol = (c & 31) * 4;           // float index within row
        int grow = row0 + trow;
        if (grow >= N) grow = N - 1;       // clamp (outputs guarded later)
        const float* gp = X + (size_t)grow * DIM + tcol;
        float* lp = Atile + trow * DIMP + tcol;
#if HAVE_ASYNC_LDS
        // AS-changing casts must go through integers; a generic pointer to
        // LDS carries the LDS byte offset in its low 32 bits (ISA aperture).
        __builtin_amdgcn_global_load_async_to_lds_b128(
            (gv4i*)(unsigned long long)(uintptr_t)gp,
            (lv4i*)(unsigned int)(uintptr_t)lp, 0, 0);
#else
        *(float4*)lp = *(const float4*)gp;
#endif
    }
#if HAVE_ASYNC_LDS
#if __has_builtin(__builtin_amdgcn_s_wait_asynccnt)
    __builtin_amdgcn_s_wait_asynccnt(0);
#endif
#endif
    __syncthreads();

    // A fragments now come from LDS (ds_load_b64, conflict-free via DIMP)
    const float* A0 = Atile + lm * DIMP + 2 * hi;          // rows 0..15
    const float* A1 = Atile + (16 + lm) * DIMP + 2 * hi;   // rows 16..31
    // B[k][n] = W[n][k] -> W[(col0+lm)*128 + k + 2*hi]
    const float* Bp = W + (size_t)(col0 + lm) * DIM + 2 * hi;

    const float bval = bias[col0 + lm];
    v8f c0, c1;
#pragma unroll
    for (int r = 0; r < 8; ++r) { c0[r] = bval; c1[r] = bval; }

#pragma unroll
    for (int k = 0; k < DIM; k += 4) {
        v2f b  = *(const v2f*)(Bp + k);
        v2f a0 = *(const v2f*)(A0 + k);
        v2f a1 = *(const v2f*)(A1 + k);
        c0 = __builtin_amdgcn_wmma_f32_16x16x4_f32(false, a0, false, b,
                                                   (short)0, c0, false, false);
        c1 = __builtin_amdgcn_wmma_f32_16x16x4_f32(false, a1, false, b,
                                                   (short)0, c1, false, false);
    }

    // C/D layout: VGPR r holds M=r (lanes 0-15) and M=r+8 (lanes 16-31), N=lm
#pragma unroll
    for (int r = 0; r < 8; ++r) {
        int row = row0 + r + 8 * hi;
        if (row < N) H[(size_t)row * DIM + col0 + lm] = c0[r];
    }
#pragma unroll
    for (int r = 0; r < 8; ++r) {
        int row = row0 + 16 + r + 8 * hi;
        if (row < N) H[(size_t)row * DIM + col0 + lm] = c1[r];
    }
}

// ---------------- edge message + segment-max ----------------
// One wave per edge; lane handles 4 contiguous channels. Native float
// atomic max (gfx1250 GLOBAL_ATOMIC_MAX_NUM_F32) at agent scope.
__device__ __forceinline__ void amaxf(float* addr, float val) {
    __hip_atomic_fetch_max(addr, val, __ATOMIC_RELAXED,
                           __HIP_MEMORY_SCOPE_AGENT);
}

__global__ void __launch_bounds__(256)
edge_kernel(const int* __restrict__ src, const int* __restrict__ dst,
            const float* __restrict__ Hsrc, const float* __restrict__ Hdst,
            const float* __restrict__ eself, float* __restrict__ agg, int E) {
    int gid = blockIdx.x * blockDim.x + threadIdx.x;
    int e = gid >> 5, lane = gid & 31;
    if (e >= E) return;
    int s = src[e], t = dst[e];
    int d = lane * 4;
    float4 hd = *(const float4*)(Hdst + (size_t)t * DIM + d);
    float4 hs = *(const float4*)(Hsrc + (size_t)s * DIM + d);
    float4 se = *(const float4*)(eself + (s == t ? DIM : 0) + d);
    float* ap = agg + (size_t)t * DIM + d;
    amaxf(ap + 0, hd.x + hs.x + se.x);
    amaxf(ap + 1, hd.y + hs.y + se.y);
    amaxf(ap + 2, hd.z + hs.z + se.z);
    amaxf(ap + 3, hd.w + hs.w + se.w);
}

// ---------------- decoder: logits + softmax (wave per row) ----------------
__global__ void __launch_bounds__(256)
decode_kernel(const float* __restrict__ X, const float* __restrict__ Wd,
              const float* __restrict__ bd, float* __restrict__ out, int N) {
    int gid = blockIdx.x * blockDim.x + threadIdx.x;
    int row = gid >> 5;
    int lane = gid & 31;
    if (row >= N) return;
    const float4 x  = *(const float4*)(X + (size_t)row * DIM + lane * 4);
    const float4 w0 = *(const float4*)(Wd + 0 * DIM + lane * 4);
    const float4 w1 = *(const float4*)(Wd + 1 * DIM + lane * 4);
    const float4 w2 = *(const float4*)(Wd + 2 * DIM + lane * 4);
    float a0 = x.x * w0.x + x.y * w0.y + x.z * w0.z + x.w * w0.w;
    float a1 = x.x * w1.x + x.y * w1.y + x.z * w1.z + x.w * w1.w;
    float a2 = x.x * w2.x + x.y * w2.y + x.z * w2.z + x.w * w2.w;
#pragma unroll
    for (int off = 16; off > 0; off >>= 1) {
        a0 += __shfl_xor(a0, off, 32);
        a1 += __shfl_xor(a1, off, 32);
        a2 += __shfl_xor(a2, off, 32);
    }
    if (lane == 0) {
        float l0 = a0 + bd[0], l1 = a1 + bd[1], l2 = a2 + bd[2];
        float* last = out + (size_t)row * NC;
        last[0] = l0; last[1] = l1; last[2] = l2;
        float mx = fmaxf(l0, fmaxf(l1, l2));
        float e0 = expf(l0 - mx), e1 = expf(l1 - mx), e2 = expf(l2 - mx);
        float inv = 1.0f / (e0 + e1 + e2);
        float* sm = out + (size_t)N * NC + (size_t)row * NC;
        sm[0] = e0 * inv; sm[1] = e1 * inv; sm[2] = e2 * inv;
    }
}

// ---------------- host orchestration ----------------
extern "C" void kernel_launch(void* const* d_in, const int* in_sizes, int n_in,
                              void* d_out, int out_size, void* d_ws,
                              size_t ws_size, hipStream_t stream) {
    const int* x0_ids = (const int*)d_in[0];
    const int* x1_ids = (const int*)d_in[1];
    const int* es[NTYPES] = {(const int*)d_in[2], (const int*)d_in[3],
                             (const int*)d_in[4], (const int*)d_in[5]};
    const float* enc_al_w   = (const float*)d_in[6];
    const float* emb_test_w = (const float*)d_in[7];
    const float* W_tgt = (const float*)d_in[8];
    const float* b_tgt = (const float*)d_in[9];
    const float* W_src = (const float*)d_in[10];
    const float* b_src = (const float*)d_in[11];
    const float* emb_self = (const float*)d_in[12];
    const float* W_dec = (const float*)d_in[13];
    const float* b_dec = (const float*)d_in[14];
    float* out = (float*)d_out;

    static const int T_SRCS[NTYPES] = {0, 0, 1, 1};
    static const int T_TGTS[NTYPES] = {0, 1, 0, 1};
    const int NODE_COUNTS[2] = {N0, N1};

    // workspace carve-up (256B aligned)
    char* base = (char*)d_ws;
    size_t off = 0;
    auto carve = [&](size_t bytes) -> float* {
        float* p = (float*)(base + off);
        off = (off + bytes + 255) & ~(size_t)255;
        return p;
    };
    const size_t B0 = (size_t)N0 * DIM * sizeof(float);
    const size_t B1 = (size_t)N1 * DIM * sizeof(float);
    float* xs0  = carve(B0);
    float* out0 = carve(B0);
    float* xs1  = carve(B1);
    float* out1 = carve(B1);
    float* hsrc = carve(B0);   // sized for the larger node type
    float* hdst = carve(B0);
    float* agg  = carve(B0);

    const int TPB = 256;
    const float NEG_INF = -__builtin_huge_valf();

    // embeddings (wave per row)
    gather_kernel<<<(N0 * 32 + TPB - 1) / TPB, TPB, 0, stream>>>(
        enc_al_w, x0_ids, xs0, N0);
    gather_kernel<<<(N1 * 32 + TPB - 1) / TPB, TPB, 0, stream>>>(
        emb_test_w, x1_ids, xs1, N1);

    float* xs[2]   = {xs0, xs1};
    float* outs[2] = {out0, out1};

    for (int i = 0; i < NLAYERS; ++i) {
        for (int j = 0; j < NTYPES; ++j) {
            const int ts = T_SRCS[j], tt = T_TGTS[j];
            const int ns = NODE_COUNTS[ts], nt = NODE_COUNTS[tt];
            const int ij = i * NTYPES + j;
            const int n4 = nt * DIM / 4;

            fill4_kernel<<<(n4 + TPB - 1) / TPB, TPB, 0, stream>>>(
                (float4*)agg, NEG_INF, n4);

            gemm_bias_wmma_kernel<<<(ns + 31) / 32, TPB, 0, stream>>>(
                xs[ts], W_src + (size_t)ij * DIM * DIM, b_src + (size_t)ij * DIM,
                hsrc, ns);
            gemm_bias_wmma_kernel<<<(nt + 31) / 32, TPB, 0, stream>>>(
                xs[tt], W_tgt + (size_t)ij * DIM * DIM, b_tgt + (size_t)ij * DIM,
                hdst, nt);

            edge_kernel<<<(NEDGE * 32 + TPB - 1) / TPB, TPB, 0, stream>>>(
                es[j], es[j] + NEDGE, hsrc, hdst,
                emb_self + (size_t)ij * 2 * DIM, agg, NEDGE);

            // j<2 is the first writer for its target type (j0->tt0, j1->tt1)
            relu_acc_kernel<<<(n4 + TPB - 1) / TPB, TPB, 0, stream>>>(
                (float4*)outs[tt], (const float4*)agg, n4, (j >= 2) ? 1 : 0);
        }
        // out = sum of relu() terms >= 0, so the reference's final relu is
        // the identity here; just swap buffer roles.
        float* t0 = xs[0]; xs[0] = outs[0]; outs[0] = t0;
        float* t1 = xs[1]; xs[1] = outs[1]; outs[1] = t1;
    }

    decode_kernel<<<((size_t)N0 * 32 + TPB - 1) / TPB, TPB, 0, stream>>>(
        xs[0], W_dec, b_dec, out, N0);
}